// AttentionLAP_72756745994553
// MI455X (gfx1250) — compile-verified
//
#include <hip/hip_runtime.h>
#include <cstdint>

// Greedy attention-LAP: B=128 batches, each a sequential 512-row masked-softmax
// scan over 512 columns. One wave32 per batch => all reductions are shuffle
// butterflies (no barriers); next row is prefetched via the CDNA5 async
// global->LDS data mover and fenced with s_wait_asynccnt (per-wave counter).

#define NEG_LOGIT (-1e30f)

constexpr int N1 = 512;           // rows per batch
constexpr int N2 = 512;           // columns per row
constexpr int K  = N2 / 32;       // 16 columns per lane (c = lane + 32k)

// Copy one 2KB row (512 f32) global -> LDS with 4 async b128 transfers.
// Each lane moves 16B; offset applies to both global and LDS addresses.
__device__ __forceinline__ void async_row_load(const float* __restrict__ gsrc,
                                               uint32_t lds_base, int lane) {
  uint64_t ga = (uint64_t)(uintptr_t)(gsrc + lane * 4);   // lane*16 bytes
  uint32_t la = lds_base + (uint32_t)(lane * 16);
  asm volatile("global_load_async_to_lds_b128 %0, %1, off"
               :: "v"(la), "v"(ga) : "memory");
  asm volatile("global_load_async_to_lds_b128 %0, %1, off offset:512"
               :: "v"(la), "v"(ga) : "memory");
  asm volatile("global_load_async_to_lds_b128 %0, %1, off offset:1024"
               :: "v"(la), "v"(ga) : "memory");
  asm volatile("global_load_async_to_lds_b128 %0, %1, off offset:1536"
               :: "v"(la), "v"(ga) : "memory");
}

__global__ __launch_bounds__(32)
void attn_lap_kernel(const float* __restrict__ s,
                     const int*   __restrict__ nrows,
                     const int*   __restrict__ ncols,
                     float*       __restrict__ out) {
  __shared__ float rowbuf[2][N2];   // double-buffered row stage (4KB LDS)

  const int b    = blockIdx.x;
  const int lane = threadIdx.x;     // wave32: lane 0..31

  const int nr = nrows[b];          // valid rows for this batch
  const int nc = ncols[b];          // allowed columns (== N2 in setup)

  const size_t batch_base = (size_t)b * N1 * N2;
  const float* sb = s   + batch_base;
  float*       ob = out + batch_base;

  // Per-lane mask register: bit k <-> column (lane + 32k) still allowed.
  uint32_t mask = 0u;
#pragma unroll
  for (int k = 0; k < K; ++k)
    if (lane + (k << 5) < nc) mask |= (1u << k);

  const uint32_t lds0 = (uint32_t)(uintptr_t)&rowbuf[0][0];
  const uint32_t lds1 = (uint32_t)(uintptr_t)&rowbuf[1][0];

  if (nr > 0) async_row_load(sb, lds0, lane);   // prologue: stage row 0

  for (int r = 0; r < nr; ++r) {
    // Row r's async copy was issued last iteration (or in the prologue) and
    // had a full row of compute to land; drain the per-wave ASYNC counter.
    asm volatile("s_wait_asynccnt 0x0" ::: "memory");

    // Kick off row r+1 into the other buffer; overlaps with compute below.
    if (r + 1 < nr)
      async_row_load(sb + (size_t)(r + 1) * N2,
                     ((r + 1) & 1) ? lds1 : lds0, lane);

    const float* buf = rowbuf[r & 1];

    // Masked logits + fused max/argmax (softmax is monotone, so
    // argmax(probs) == argmax(masked logits); ties -> lowest index,
    // matching jnp.argmax).
    float v[K];
    float bestv = NEG_LOGIT;
    int   besti = N2;
#pragma unroll
    for (int k = 0; k < K; ++k) {
      const int   c = lane + (k << 5);
      const bool  m = (mask >> k) & 1u;
      const float x = buf[c];                 // stride-4B: bank-conflict-free
      const float val = m ? x : NEG_LOGIT;
      v[k] = val;
      if (val > bestv || (val == bestv && c < besti)) { bestv = val; besti = c; }
    }
    // Wave32 butterfly reduce; every lane converges to the same (max, idx).
#pragma unroll
    for (int off = 16; off; off >>= 1) {
      const float ov = __shfl_xor(bestv, off, 32);
      const int   oi = __shfl_xor(besti, off, 32);
      if (ov > bestv || (ov == bestv && oi < besti)) { bestv = ov; besti = oi; }
    }

    // exp(logit - max), zero on masked cols, butterfly-sum the denominator.
    float p[K];
    float acc = 0.0f;
#pragma unroll
    for (int k = 0; k < K; ++k) {
      const bool m = (mask >> k) & 1u;
      const float e = m ? __expf(v[k] - bestv) : 0.0f;   // v_exp_f32
      p[k] = e;
      acc += e;
    }
#pragma unroll
    for (int off = 16; off; off >>= 1)
      acc += __shfl_xor(acc, off, 32);

    const float inv = 1.0f / acc;
    float* orow = ob + (size_t)r * N2;
#pragma unroll
    for (int k = 0; k < K; ++k)
      orow[lane + (k << 5)] = p[k] * inv;     // 128B-coalesced b32 stores

    // Greedy removal: clear the winning column's mask bit (active rows only;
    // we only iterate active rows here).
    if ((besti & 31) == lane) mask &= ~(1u << (besti >> 5));
  }

  // Inactive rows (r >= nrows[b]) are exactly zero in the reference.
  for (int r = nr; r < N1; ++r) {
    float4* orow4 = (float4*)(ob + (size_t)r * N2);
    const float4 z = {0.0f, 0.0f, 0.0f, 0.0f};
#pragma unroll
    for (int k = 0; k < 4; ++k)
      orow4[lane + (k << 5)] = z;             // b128 zero-fill
  }
}

extern "C" void kernel_launch(void* const* d_in, const int* in_sizes, int n_in,
                              void* d_out, int out_size, void* d_ws, size_t ws_size,
                              hipStream_t stream) {
  (void)n_in; (void)out_size; (void)d_ws; (void)ws_size;
  const float* s     = (const float*)d_in[0];
  const int*   nrows = (const int*)d_in[1];
  const int*   ncols = (const int*)d_in[2];
  float*       out   = (float*)d_out;

  const int B = in_sizes[1];                  // 128 batches, one wave32 each
  attn_lap_kernel<<<B, 32, 0, stream>>>(s, nrows, ncols, out);
}